// SkipGramModel_2542620640014
// MI455X (gfx1250) — compile-verified
//
#include <hip/hip_runtime.h>

typedef float v2f __attribute__((ext_vector_type(2)));
typedef float v8f __attribute__((ext_vector_type(8)));

#define VOCAB 100000
#define EMBED 128
#define BATCH 65536
#define NEG   20

#define WAVES_PER_BLOCK 8
#define TILE 16                                   // batch elements per wave
#define BLOCK_THREADS (WAVES_PER_BLOCK * 32)      // 256
#define ELEMS_PER_BLOCK (WAVES_PER_BLOCK * TILE)  // 128
#define GRID1 (BATCH / ELEMS_PER_BLOCK)           // 512

__device__ __forceinline__ float logsig(float x) {
  // log(sigmoid(x)) = min(x,0) - log1p(exp(-|x|)), numerically stable
  return fminf(x, 0.0f) - log1pf(__expf(-fabsf(x)));
}

__global__ void __launch_bounds__(BLOCK_THREADS)
sgns_partials(const int* __restrict__ tgt, const int* __restrict__ ctx,
              const int* __restrict__ neg, const float* __restrict__ Wh,
              const float* __restrict__ Wo, float* __restrict__ partials) {
  const int lane  = threadIdx.x & 31;
  const int wave  = threadIdx.x >> 5;
  const int tile  = (blockIdx.x * WAVES_PER_BLOCK + wave) * TILE;
  const int b     = tile + (lane & 15);   // batch element this lane serves (row M = lane&15)
  const int kbase = (lane >> 4) * 2;      // K sub-offset inside a 4-wide slice (A/B VGPR layout)

  // Row offsets (f32 elements). Max 100000*128 = 12.8M, fits int.
  const int toff = tgt[b] * EMBED;
  const int coff = ctx[b] * EMBED;
  int nof[NEG];
#pragma unroll
  for (int k = 0; k < NEG; ++k) nof[k] = neg[b * NEG + k] * EMBED;

  v8f cpos = {};   // 16x16 f32 accumulator: diag = pos scores for tile
  v8f cneg = {};   // diag = summed-neg scores for tile

#pragma unroll 2
  for (int kk = 0; kk < EMBED / 4; ++kk) {
    const int ko = kk * 4 + kbase;
    // A-matrix fragment: Eh[tile .. tile+15, kk*4 .. kk*4+3]
    v2f a  = *(const v2f*)(Wh + toff + ko);
    // B-matrix fragment (pos): Eo[ctx] column layout, same address pattern
    v2f bm = *(const v2f*)(Wo + coff + ko);
    // Elementwise sum of the 20 negative embeddings (valid: logsigmoid applied after sum over k)
    v2f n;
    n.x = 0.0f; n.y = 0.0f;
#pragma unroll
    for (int k = 0; k < NEG; ++k) {
      v2f t = *(const v2f*)(Wo + nof[k] + ko);
      n.x += t.x;
      n.y += t.y;
    }
    // D = A*B + C, f32 16x16x4  (8 args: neg_a, A, neg_b, B, c_mod, C, reuse_a, reuse_b)
    cpos = __builtin_amdgcn_wmma_f32_16x16x4_f32(false, a, false, bm, (short)0, cpos, false, false);
    cneg = __builtin_amdgcn_wmma_f32_16x16x4_f32(false, a, false, n,  (short)0, cneg, false, false);
  }

  // Extract diagonals: (m,m) for m<8 lives in VGPR m, lane m; for m>=8 in VGPR m-8, lane m+16.
  float s = 0.0f;
#pragma unroll
  for (int m = 0; m < 8; ++m) {
    float p = __shfl(cpos[m], m, 32);        // element (m, m)
    float q = __shfl(cneg[m], m, 32);
    s += logsig(p) + logsig(-q);
    p = __shfl(cpos[m], m + 24, 32);         // element (m+8, m+8): lane (m+8)+16
    q = __shfl(cneg[m], m + 24, 32);
    s += logsig(p) + logsig(-q);
  }

  __shared__ float wsum[WAVES_PER_BLOCK];
  if (lane == 0) wsum[wave] = s;
  __syncthreads();
  if (threadIdx.x == 0) {
    float t = 0.0f;
#pragma unroll
    for (int i = 0; i < WAVES_PER_BLOCK; ++i) t += wsum[i];
    partials[blockIdx.x] = t;
  }
}

__global__ void __launch_bounds__(256)
sgns_reduce(const float* __restrict__ partials, float* __restrict__ out, int n) {
  __shared__ float buf[256];
  float t = 0.0f;
  for (int i = threadIdx.x; i < n; i += 256) t += partials[i];
  buf[threadIdx.x] = t;
  __syncthreads();
  for (int s = 128; s > 0; s >>= 1) {
    if ((int)threadIdx.x < s) buf[threadIdx.x] += buf[threadIdx.x + s];
    __syncthreads();
  }
  if (threadIdx.x == 0) out[0] = -buf[0];   // loss = -(pos_loss + neg_loss)
}

extern "C" void kernel_launch(void* const* d_in, const int* in_sizes, int n_in,
                              void* d_out, int out_size, void* d_ws, size_t ws_size,
                              hipStream_t stream) {
  const int*   tgt = (const int*)d_in[0];
  const int*   ctx = (const int*)d_in[1];
  const int*   neg = (const int*)d_in[2];
  const float* Wh  = (const float*)d_in[3];
  const float* Wo  = (const float*)d_in[4];
  float* partials  = (float*)d_ws;          // GRID1 floats = 2 KB scratch

  sgns_partials<<<GRID1, BLOCK_THREADS, 0, stream>>>(tgt, ctx, neg, Wh, Wo, partials);
  sgns_reduce<<<1, 256, 0, stream>>>(partials, (float*)d_out, GRID1);

  (void)in_sizes; (void)n_in; (void)out_size; (void)ws_size;
}